// WmDeformer_69569880261285
// MI455X (gfx1250) — compile-verified
//
#include <hip/hip_runtime.h>
#include <hip/hip_bf16.h>
#include <math.h>

#define NODES_F 512
#define HEADS 6
#define CF 256
#define CC 2
#define XPC_STRIDE 16   // padded coord projection row stride (>= HEADS*CC = 12)

typedef __attribute__((ext_vector_type(2))) float v2f;
typedef __attribute__((ext_vector_type(8))) float v8f;

__device__ __forceinline__ float lrelu(float x) { return x > 0.f ? x : 0.2f * x; }
__device__ __forceinline__ float selu_f(float x) {
    const float sc = 1.0507009873554805f, al = 1.6732632423543772f;
    return x > 0.f ? sc * x : sc * al * (expf(x) - 1.f);
}

// ---------------------------------------------------------------------------
// GEMM: C[M,NP] = X[M,K] * W[K,NP] with V_WMMA_F32_16X16X4_F32.
// NP is a compile-time stride (multiple of 16*SUB) so all B/C accesses use
// immediate offsets from one base pointer -> clause-able loads, no masks.
// One wave computes a 16 x (16*SUB) strip; the A fragment is reused by SUB
// WMMAs per k-step.
// ---------------------------------------------------------------------------
template <int NP, int SUB>
__global__ void gat_gemm_wmma_f32(const float* __restrict__ X,
                                  const float* __restrict__ W,
                                  float* __restrict__ C,
                                  int M, int K) {
    const int lane = threadIdx.x & 31;
    const int wave = threadIdx.x >> 5;
    const int mt = M >> 4;                        // M tiles
    const int ns = NP / (16 * SUB);               // N strips
    const int strips = mt * ns;
    int strip = blockIdx.x * (blockDim.x >> 5) + wave;
    if (strip >= strips) return;                  // uniform per wave
    const int tm = strip % mt;
    const int sn = strip / mt;

    const int l16 = lane & 15;
    const int khalf = (lane >> 4) * 2;            // K offset (0 or 2) in group of 4

    const float* ap  = X + (size_t)(tm * 16 + l16) * K + khalf;      // A fragment
    const float* wpb = W + (size_t)khalf * NP + sn * (16 * SUB) + l16;

    v8f acc[SUB];
#pragma unroll
    for (int s = 0; s < SUB; ++s) acc[s] = (v8f){};

#pragma unroll 4
    for (int k0 = 0; k0 < K; k0 += 4) {
        v2f a = *(const v2f*)ap;                  // 8B aligned (khalf even)
#pragma unroll
        for (int s = 0; s < SUB; ++s) {
            v2f b;
            b.x = wpb[s * 16];                    // row khalf,   col s*16+l16
            b.y = wpb[s * 16 + NP];               // row khalf+1, col s*16+l16
            acc[s] = __builtin_amdgcn_wmma_f32_16x16x4_f32(false, a, false, b,
                                                           (short)0, acc[s],
                                                           false, false);
        }
        ap  += 4;
        wpb += (size_t)4 * NP;
    }

    // C/D layout: VGPR r -> row tm*16 + r + 8*(lane>>4), col sn*16*SUB + s*16 + l16
    float* cp = C + (size_t)(tm * 16 + 8 * (lane >> 4)) * NP + sn * (16 * SUB) + l16;
#pragma unroll
    for (int s = 0; s < SUB; ++s)
#pragma unroll
        for (int r = 0; r < 8; ++r)
            cp[(size_t)r * NP + s * 16] = acc[s][r];
}

// Pad W_coord [K,12] -> Wpad [K,16] (zero-filled tail columns).
__global__ void gat_pad_wc(const float* __restrict__ Wc, float* __restrict__ Wpad,
                           int K) {
    int idx = blockIdx.x * blockDim.x + threadIdx.x;
    if (idx >= K * XPC_STRIDE) return;
    int row = idx / XPC_STRIDE, col = idx % XPC_STRIDE;
    Wpad[idx] = (col < HEADS * CC) ? Wc[row * (HEADS * CC) + col] : 0.f;
}

// ---------------------------------------------------------------------------
// a_src[n,h] = sum_c xp[n,h,c]*att_src[h,c];  a_dst likewise.
// xp row stride passed separately (padded for coord head).
// ---------------------------------------------------------------------------
__global__ void gat_attn_proj(const float* __restrict__ xp,
                              const float* __restrict__ att_src,
                              const float* __restrict__ att_dst,
                              float* __restrict__ asrc, float* __restrict__ adst,
                              int n, int C, int stride) {
    int idx = blockIdx.x * blockDim.x + threadIdx.x;
    if (idx >= n * HEADS) return;
    int node = idx / HEADS, h = idx % HEADS;
    const float* row = xp + (size_t)node * stride + (size_t)h * C;
    const float* as = att_src + h * C;
    const float* ad = att_dst + h * C;
    float s0 = 0.f, s1 = 0.f;
    for (int c = 0; c < C; ++c) { float v = row[c]; s0 += v * as[c]; s1 += v * ad[c]; }
    asrc[idx] = s0; adst[idx] = s1;
}

// ---------------------------------------------------------------------------
// CSR-by-dst construction (counting sort). Self-loops are edges [E, E+n).
// ---------------------------------------------------------------------------
__global__ void gat_degree(const int* __restrict__ ei, int E, int n,
                           int* __restrict__ deg) {
    int e = blockIdx.x * blockDim.x + threadIdx.x;
    int Et = E + n;
    if (e >= Et) return;
    int d = (e < E) ? ei[E + e] : (e - E);
    atomicAdd(&deg[d], 1);
}

__global__ void gat_scan(const int* __restrict__ deg, int* __restrict__ rowstart, int n) {
    __shared__ int buf[1024];
    __shared__ int carry;
    if (threadIdx.x == 0) carry = 0;
    __syncthreads();
    for (int base = 0; base < n; base += 1024) {
        int i = base + (int)threadIdx.x;
        int v = (i < n) ? deg[i] : 0;
        buf[threadIdx.x] = v;
        __syncthreads();
        for (int off = 1; off < 1024; off <<= 1) {
            int t = (threadIdx.x >= (unsigned)off) ? buf[threadIdx.x - off] : 0;
            __syncthreads();
            buf[threadIdx.x] += t;
            __syncthreads();
        }
        if (i < n) rowstart[i] = carry + buf[threadIdx.x] - v;  // exclusive
        __syncthreads();
        if (threadIdx.x == 1023) carry += buf[1023];
        __syncthreads();
    }
}

__global__ void gat_scatter(const int* __restrict__ ei, int E, int n,
                            const int* __restrict__ rowstart,
                            int* __restrict__ cursor, int* __restrict__ csr_src) {
    int e = blockIdx.x * blockDim.x + threadIdx.x;
    int Et = E + n;
    if (e >= Et) return;
    int s, d;
    if (e < E) { s = ei[e]; d = ei[E + e]; } else { s = e - E; d = e - E; }
    int pos = rowstart[d] + atomicAdd(&cursor[d], 1);
    csr_src[pos] = s;
}

// ---------------------------------------------------------------------------
// Feat aggregation: one 256-thread block per node; thread t owns output
// channel t across all 6 heads. Softmax coefs staged once per block in LDS.
// ---------------------------------------------------------------------------
#define AGG_CHUNK 32
__global__ void gat_agg_feat(const float* __restrict__ xp,      // [n, 1536]
                             const float* __restrict__ asrc,    // [n, 6]
                             const float* __restrict__ adst,    // [n, 6]
                             const int* __restrict__ rowstart,
                             const int* __restrict__ deg,
                             const int* __restrict__ csr_src,
                             const float* __restrict__ bias,    // [256]
                             float* __restrict__ out, int n) {
    __shared__ float s_amax[HEADS];
    __shared__ float s_denom[HEADS];
    __shared__ float s_coef[AGG_CHUNK][HEADS];
    __shared__ int   s_src[AGG_CHUNK];

    const int node = blockIdx.x;
    const int t = threadIdx.x;
    const int start = rowstart[node];
    const int d = deg[node];

    if (t < HEADS) {
        const int h = t;
        const float ad = adst[node * HEADS + h];
        float m = -INFINITY;
        for (int e = 0; e < d; ++e) {
            int s = csr_src[start + e];
            m = fmaxf(m, lrelu(asrc[s * HEADS + h] + ad));
        }
        float dn = 0.f;
        for (int e = 0; e < d; ++e) {
            int s = csr_src[start + e];
            dn += expf(lrelu(asrc[s * HEADS + h] + ad) - m);
        }
        s_amax[h] = m;
        s_denom[h] = dn + 1e-16f;
    }
    __syncthreads();

    float acc = 0.f;
    for (int c0 = 0; c0 < d; c0 += AGG_CHUNK) {
        const int cn = min(AGG_CHUNK, d - c0);
        if (t < cn) s_src[t] = csr_src[start + c0 + t];
        __syncthreads();
        if (t < cn * HEADS) {
            const int el = t / HEADS, h = t % HEADS;
            const int s = s_src[el];
            float al = lrelu(asrc[s * HEADS + h] + adst[node * HEADS + h]);
            s_coef[el][h] = expf(al - s_amax[h]) / s_denom[h];
        }
        __syncthreads();
        for (int el = 0; el < cn; ++el) {
            const int s = s_src[el];
            const float* row = xp + (size_t)s * (HEADS * CF) + t;
#pragma unroll
            for (int h = 0; h < HEADS; ++h)
                acc += s_coef[el][h] * row[h * CF];
        }
        __syncthreads();
    }
    float v = acc * (1.f / (float)HEADS) + bias[t];
    out[(size_t)node * CF + t] = selu_f(v);
}

// ---------------------------------------------------------------------------
// Coord aggregation (C=2, padded stride 16): one thread per node + boundary fix.
// ---------------------------------------------------------------------------
__global__ void gat_agg_coord(const float* __restrict__ xp,     // [n, 16] padded
                              const float* __restrict__ asrc,
                              const float* __restrict__ adst,
                              const int* __restrict__ rowstart,
                              const int* __restrict__ deg,
                              const int* __restrict__ csr_src,
                              const float* __restrict__ bias,   // [2]
                              const float* __restrict__ data,   // [n, 512]
                              float* __restrict__ out, int n) {
    int node = blockIdx.x * blockDim.x + threadIdx.x;
    if (node >= n) return;
    const int start = rowstart[node];
    const int d = deg[node];
    float ad[HEADS], amax[HEADS], denom[HEADS];
#pragma unroll
    for (int h = 0; h < HEADS; ++h) { ad[h] = adst[node * HEADS + h]; amax[h] = -INFINITY; }
    for (int e = 0; e < d; ++e) {
        int s = csr_src[start + e];
#pragma unroll
        for (int h = 0; h < HEADS; ++h)
            amax[h] = fmaxf(amax[h], lrelu(asrc[s * HEADS + h] + ad[h]));
    }
#pragma unroll
    for (int h = 0; h < HEADS; ++h) denom[h] = 0.f;
    for (int e = 0; e < d; ++e) {
        int s = csr_src[start + e];
#pragma unroll
        for (int h = 0; h < HEADS; ++h)
            denom[h] += expf(lrelu(asrc[s * HEADS + h] + ad[h]) - amax[h]);
    }
    float a0 = 0.f, a1 = 0.f;
    for (int e = 0; e < d; ++e) {
        int s = csr_src[start + e];
        const float* row = xp + (size_t)s * XPC_STRIDE;
#pragma unroll
        for (int h = 0; h < HEADS; ++h) {
            float coef = expf(lrelu(asrc[s * HEADS + h] + ad[h]) - amax[h]) /
                         (denom[h] + 1e-16f);
            a0 += coef * row[h * CC + 0];
            a1 += coef * row[h * CC + 1];
        }
    }
    float c0 = a0 * (1.f / (float)HEADS) + bias[0];
    float c1 = a1 * (1.f / (float)HEADS) + bias[1];
    const float d0 = data[(size_t)node * NODES_F + 0];
    const float d1 = data[(size_t)node * NODES_F + 1];
    if (d0 == 1.0f) c0 = 1.0f;
    if (d0 == 0.0f) c0 = 0.0f;
    if (d1 == 0.0f) c1 = 0.0f;
    if (d1 == 1.0f) c1 = 1.0f;
    out[(size_t)node * CC + 0] = c0;
    out[(size_t)node * CC + 1] = c1;
}

// ---------------------------------------------------------------------------
extern "C" void kernel_launch(void* const* d_in, const int* in_sizes, int n_in,
                              void* d_out, int out_size, void* d_ws, size_t ws_size,
                              hipStream_t stream) {
    const float* data   = (const float*)d_in[0];
    const int*   ei     = (const int*)d_in[1];
    const float* Wc     = (const float*)d_in[2];
    const float* asc    = (const float*)d_in[3];
    const float* adc    = (const float*)d_in[4];
    const float* bc     = (const float*)d_in[5];
    const float* Wf     = (const float*)d_in[6];
    const float* asf    = (const float*)d_in[7];
    const float* adf    = (const float*)d_in[8];
    const float* bf     = (const float*)d_in[9];

    const int Nn = in_sizes[0] / NODES_F;   // 20000
    const int E  = in_sizes[1] / 2;         // 160000
    const int Et = E + Nn;
    const int K  = NODES_F;                 // 512
    const int NF = HEADS * CF;              // 1536

    float* out_coord = (float*)d_out;                     // [Nn,2]
    float* out_feat  = (float*)d_out + (size_t)Nn * CC;   // [Nn,256]

    // ---- workspace carve-up (all 4B types, base is 256B aligned) ----
    char* p = (char*)d_ws;
    float* xp_feat  = (float*)p; p += (size_t)Nn * NF * 4;
    float* xp_coord = (float*)p; p += (size_t)Nn * XPC_STRIDE * 4;  // padded
    float* wc_pad   = (float*)p; p += (size_t)K * XPC_STRIDE * 4;   // padded W_coord
    float* asrc_f   = (float*)p; p += (size_t)Nn * HEADS * 4;
    float* adst_f   = (float*)p; p += (size_t)Nn * HEADS * 4;
    float* asrc_c   = (float*)p; p += (size_t)Nn * HEADS * 4;
    float* adst_c   = (float*)p; p += (size_t)Nn * HEADS * 4;
    int* deg        = (int*)p;   p += (size_t)Nn * 4;
    int* rowstart   = (int*)p;   p += (size_t)Nn * 4;
    int* cursor     = (int*)p;   p += (size_t)Nn * 4;
    int* csr_src    = (int*)p;   p += (size_t)Et * 4;
    (void)ws_size; (void)n_in; (void)out_size;

    // ---- 1) projections via f32 WMMA ----
    {   // feat: [Nn,512] x [512,1536], 16x64 strips per wave
        int strips = (Nn / 16) * (NF / 64);
        int blocks = (strips + 7) / 8;   // 8 waves / 256-thread block
        gat_gemm_wmma_f32<NF, 4><<<blocks, 256, 0, stream>>>(data, Wf, xp_feat, Nn, K);
    }
    {   // coord: pad W to [512,16], then [Nn,512] x [512,16]
        int padel = K * XPC_STRIDE;
        gat_pad_wc<<<(padel + 255) / 256, 256, 0, stream>>>(Wc, wc_pad, K);
        int strips = (Nn / 16) * (XPC_STRIDE / 16);
        int blocks = (strips + 7) / 8;
        gat_gemm_wmma_f32<XPC_STRIDE, 1><<<blocks, 256, 0, stream>>>(data, wc_pad,
                                                                     xp_coord, Nn, K);
    }

    // ---- 2) attention scalars ----
    {
        int th = Nn * HEADS;
        int blocks = (th + 255) / 256;
        gat_attn_proj<<<blocks, 256, 0, stream>>>(xp_feat, asf, adf, asrc_f, adst_f,
                                                  Nn, CF, NF);
        gat_attn_proj<<<blocks, 256, 0, stream>>>(xp_coord, asc, adc, asrc_c, adst_c,
                                                  Nn, CC, XPC_STRIDE);
    }

    // ---- 3) CSR by dst (counting sort) ----
    hipMemsetAsync(deg, 0, (size_t)Nn * 4, stream);
    hipMemsetAsync(cursor, 0, (size_t)Nn * 4, stream);
    {
        int blocks = (Et + 255) / 256;
        gat_degree<<<blocks, 256, 0, stream>>>(ei, E, Nn, deg);
        gat_scan<<<1, 1024, 0, stream>>>(deg, rowstart, Nn);
        gat_scatter<<<blocks, 256, 0, stream>>>(ei, E, Nn, rowstart, cursor, csr_src);
    }

    // ---- 4) per-node softmax + aggregation ----
    gat_agg_feat<<<Nn, 256, 0, stream>>>(xp_feat, asrc_f, adst_f, rowstart, deg,
                                         csr_src, bf, out_feat, Nn);
    {
        int blocks = (Nn + 255) / 256;
        gat_agg_coord<<<blocks, 256, 0, stream>>>(xp_coord, asrc_c, adst_c, rowstart, deg,
                                                  csr_src, bc, data, out_coord, Nn);
    }
}